// tokenEmbedding_25194278158588
// MI455X (gfx1250) — compile-verified
//
#include <hip/hip_runtime.h>
#include <stdint.h>

// Token embedding gather: out[t, :] = table[x[t], :]
//   x:     int32[32768]          (d_in[0])
//   table: float32[50257, 512]   (d_in[1])
//   out:   float32[32768, 512]   (d_out)
//
// Strategy: CDNA5 Tensor Data Mover in gather mode. One wave per workgroup
// gathers 16 rows (16-bit row indices, vocab < 65536) of 512 f32 each into a
// 32KB LDS tile with a single TENSOR_LOAD_TO_LDS, then streams the tile to the
// output with one contiguous TENSOR_STORE_FROM_LDS. Pure DMA data path;
// HBM traffic is the roofline minimum (64MB read + 64MB write).

typedef __attribute__((ext_vector_type(4))) unsigned int v4u;
typedef __attribute__((ext_vector_type(8))) int          v8i;
typedef __attribute__((ext_vector_type(4))) int          v4i;
typedef __attribute__((ext_vector_type(4))) float        v4f;

#define EMB_D       512      // floats per row
#define TOK_PER_WG  16       // rows per gather descriptor (16-bit index mode max)

// Kernel must be *defined* in both host and device passes so the launch stub
// resolves; only the body is device-conditional.
__global__ __launch_bounds__(32) void tokenEmbedding_tdm_kernel(
    const int* __restrict__ x, const float* __restrict__ table,
    float* __restrict__ out) {
#if defined(__HIP_DEVICE_COMPILE__)
#if __has_builtin(__builtin_amdgcn_tensor_load_to_lds) && \
    __has_builtin(__builtin_amdgcn_tensor_store_from_lds)
  // ---------------- TDM gather path ----------------
  // 16 rows x 512 f32 = 32 KB staging tile. First (only) LDS object -> offset 0.
  __shared__ float lds_tile[TOK_PER_WG * EMB_D];
  // Keep the LDS allocation alive (TDM writes it by address, compiler can't see that).
  asm volatile("" ::"v"((unsigned)(unsigned long long)(uintptr_t)lds_tile));

  const unsigned wg = blockIdx.x;
  const int* tok = x + (size_t)wg * TOK_PER_WG;

  // Pack 16 token ids as 16-bit row indices, two per dword (ISA 8.7, 16-bit mode).
  unsigned p[8];
#pragma unroll
  for (int i = 0; i < 8; ++i) {
    unsigned lo = (unsigned)tok[2 * i] & 0xFFFFu;
    unsigned hi = (unsigned)tok[2 * i + 1] & 0xFFFFu;
    p[i] = lo | (hi << 16);
  }

  // ---------------- gather-load descriptor ----------------
  // Group 0: count=1 | gather_mode(bit31) | 16-bit idx(bit30=0); lds_addr=0;
  //          global_addr = table base (bits 120:64); type=2 in bits 127:126.
  unsigned long long gsrc = (unsigned long long)(uintptr_t)table;
  v4u g0l = {0x80000001u,
             0u,
             (unsigned)(gsrc & 0xFFFFFFFFu),
             ((unsigned)((gsrc >> 32) & 0x1FFFFFFu)) | 0x80000000u};
  // Group 1 bit packing (ISA 8.4):
  //  w0: wg_mask=0, data_size=2 (4B)                     -> 0x00020000
  //  w1: atomic_barrier_addr=0 | tensor_dim0.lo16 << 16  (tensor_dim0 = 512)
  //  w2: tensor_dim0.hi16 | tensor_dim1.lo16 << 16       (tensor_dim1 = 50257)
  //  w3: tensor_dim1.hi16 | tile_dim0 << 16              (tile_dim0 = 512)
  //  w4: tile_dim1 (=#valid indices) | tile_dim2 << 16   (tile_dim2 ignored)
  //  w5: tensor_dim0_stride.lo32                         (= 512 elements)
  //  w6,w7: stride0.hi16 / dim1_stride (ignored in gather) = 0
  v8i g1l = {(int)0x00020000u,
             (int)(512u << 16),
             (int)(50257u << 16),
             (int)(512u << 16),
             (int)TOK_PER_WG,
             (int)512,
             0,
             0};
  v4i g2l = {(int)p[0], (int)p[1], (int)p[2], (int)p[3]};  // row_index 0..7
  v4i g3l = {(int)p[4], (int)p[5], (int)p[6], (int)p[7]};  // row_index 8..15

#if __clang_major__ >= 23
  v8i g4z = {0, 0, 0, 0, 0, 0, 0, 0};
  __builtin_amdgcn_tensor_load_to_lds(g0l, g1l, g2l, g3l, g4z, 0);
#else
  __builtin_amdgcn_tensor_load_to_lds(g0l, g1l, g2l, g3l, 0);
#endif
  __builtin_amdgcn_s_wait_tensorcnt((short)0);

  // ---------------- contiguous store descriptor ----------------
  // The 16 gathered rows are contiguous in the output: one 1-D tile of
  // 8192 f32 (32 KB) at out + wg*TOK_PER_WG*EMB_D.
  unsigned long long gdst =
      (unsigned long long)(uintptr_t)(out + (size_t)wg * TOK_PER_WG * EMB_D);
  v4u g0s = {0x00000001u,  // count=1, normal mode
             0u,
             (unsigned)(gdst & 0xFFFFFFFFu),
             ((unsigned)((gdst >> 32) & 0x1FFFFFFu)) | 0x80000000u};
  v8i g1s = {(int)0x00020000u,      // data_size = 4B
             (int)(8192u << 16),    // tensor_dim0 = 8192
             (int)(1u << 16),       // tensor_dim1 = 1
             (int)(8192u << 16),    // tile_dim0 = 8192
             1,                     // tile_dim1 = 1
             8192,                  // tensor_dim0_stride = 8192
             0,
             0};
  v4i z4 = {0, 0, 0, 0};

#if __clang_major__ >= 23
  __builtin_amdgcn_tensor_store_from_lds(g0s, g1s, z4, z4, g4z, 0);
#else
  __builtin_amdgcn_tensor_store_from_lds(g0s, g1s, z4, z4, 0);
#endif
  // LDS is deallocated when the wave ends; the DMA store must be complete.
  __builtin_amdgcn_s_wait_tensorcnt((short)0);

#else
  // ---------------- device fallback (no TDM builtin) ----------------
  // Same launch geometry: 32 lanes copy 16 rows (2048 float4) directly.
  const unsigned wg = blockIdx.x;
  const int* tok = x + (size_t)wg * TOK_PER_WG;
  const unsigned lane = threadIdx.x;
#pragma unroll 4
  for (unsigned i = lane; i < TOK_PER_WG * (EMB_D / 4); i += 32) {
    unsigned row = i >> 7;          // / 128
    unsigned col = i & 127;
    int t = tok[row];
    v4f v = *((const v4f*)table + (size_t)t * (EMB_D / 4) + col);
    __builtin_nontemporal_store(
        v, (v4f*)out + (size_t)wg * TOK_PER_WG * (EMB_D / 4) + i);
  }
#endif
#else
  // Host pass: empty body (only the symbol/stub is needed).
  (void)x; (void)table; (void)out;
#endif
}

extern "C" void kernel_launch(void* const* d_in, const int* in_sizes, int n_in,
                              void* d_out, int out_size, void* d_ws,
                              size_t ws_size, hipStream_t stream) {
  const int* x = (const int*)d_in[0];          // [B*L] token ids
  const float* table = (const float*)d_in[1];  // [VOCAB, 512]
  float* out = (float*)d_out;                  // [B*L, 512]
  const int n_tok = in_sizes[0];               // 32768

  const int n_wg = n_tok / TOK_PER_WG;  // 2048 workgroups, 1 wave32 each
  tokenEmbedding_tdm_kernel<<<n_wg, 32, 0, stream>>>(x, table, out);
}